// NPSparseTransitionPrior_80685255623304
// MI455X (gfx1250) — compile-verified
//
#include <hip/hip_runtime.h>
#include <hip/hip_bf16.h>
#include <math.h>

typedef float v2f __attribute__((ext_vector_type(2)));
typedef float v8f __attribute__((ext_vector_type(8)));

#define ZDIM 16
#define HDIM 64
#define BDIM 128
#define TDIM 1024
#define TW   1023
#define NTOT (BDIM*TW)      /* 130944 */
#define SLOPEV 0.2f
#define EPSV 1e-6f

/* shared-memory float offsets */
#define OW0 0                      /* W0 padded [64][20], zeros for f>=17 : 1280 */
#define OW1 1280                   /* W1 [g][h] stride 65 : 4160 */
#define OW2 (OW1+4160)             /* W2 [g][h] stride 65 : 4160 */
#define OW3 (OW2+4160)             /* 64 */
#define OB0 (OW3+64)
#define OB1 (OB0+64)
#define OB2 (OB1+64)
#define OB3 (OB2+64)
#define OSCR (OB3+2)               /* even -> 8B aligned scratch */
#define SCR_STRIDE 68
#define SCR_PER_WAVE (16*SCR_STRIDE)
#define SMEM_FLOATS (OSCR + 4*SCR_PER_WAVE)

/* output section offsets (floats) */
#define RES_OFF ((size_t)0)
#define SLD_OFF ((size_t)ZDIM*NTOT)
#define TJ_OFF  (SLD_OFF + BDIM)
#define XX_OFF  (TJ_OFF + (size_t)NTOT*256)
#define YY_OFF  (XX_OFF + TW)

__device__ __forceinline__ v8f wmma4(v2f a, v2f b, v8f c) {
  return __builtin_amdgcn_wmma_f32_16x16x4_f32(false, a, false, b, (short)0, c, false, false);
}
__device__ __forceinline__ v8f splat8(float s) {
  v8f v = {s,s,s,s,s,s,s,s}; return v;
}

/* forward 16x64 = A(16x64 in scratch) * W^T, B[k=h][col=g] = W[col*65+k] */
__device__ __forceinline__ void gemm_fwd64(const float* scr, const float* W, const float* bias,
                                           int ln, int kh2, v8f acc[4]) {
  #pragma unroll
  for (int t4 = 0; t4 < 4; ++t4) acc[t4] = splat8(bias[t4*16 + ln]);
  for (int kk = 0; kk < 16; ++kk) {
    int k0 = kk*4 + kh2;
    const float* ap = scr + ln*SCR_STRIDE + k0;
    v2f a; a.x = ap[0]; a.y = ap[1];
    #pragma unroll
    for (int t4 = 0; t4 < 4; ++t4) {
      const float* bp = W + (t4*16 + ln)*65 + k0;
      v2f b; b.x = bp[0]; b.y = bp[1];
      acc[t4] = wmma4(a, b, acc[t4]);
    }
  }
}

/* backward 16x64 = A(16x64 in scratch) * W, B[k=g][col=h] = W[k*65+col] */
__device__ __forceinline__ void gemm_bwd64(const float* scr, const float* W,
                                           int ln, int kh2, v8f acc[4]) {
  #pragma unroll
  for (int t4 = 0; t4 < 4; ++t4) acc[t4] = splat8(0.0f);
  for (int kk = 0; kk < 16; ++kk) {
    int k0 = kk*4 + kh2;
    const float* ap = scr + ln*SCR_STRIDE + k0;
    v2f a; a.x = ap[0]; a.y = ap[1];
    #pragma unroll
    for (int t4 = 0; t4 < 4; ++t4) {
      int col = t4*16 + ln;
      v2f b; b.x = W[k0*65 + col]; b.y = W[(k0+1)*65 + col];
      acc[t4] = wmma4(a, b, acc[t4]);
    }
  }
}

__global__ __launch_bounds__(128)
void mlp_jac_kernel(const float* __restrict__ x,
                    const float* __restrict__ W0, const float* __restrict__ b0,
                    const float* __restrict__ W1, const float* __restrict__ b1,
                    const float* __restrict__ W2, const float* __restrict__ b2,
                    const float* __restrict__ W3, const float* __restrict__ b3,
                    float* __restrict__ out) {
  __shared__ float sm[SMEM_FLOATS];
  const int tid  = threadIdx.x;
  const int wave = tid >> 5;
  const int lane = tid & 31;
  const int ln   = lane & 15;
  const int half = lane >> 4;
  const int kh2  = half << 1;
  const int z    = blockIdx.x / 2046;
  const int n0   = ((blockIdx.x % 2046) * 4 + wave) * 16;

  /* cooperative weight load (one z per block) */
  for (int i = tid; i < 1280; i += 128) {
    int h = i / 20, f = i % 20;
    sm[OW0 + i] = (f < 17) ? W0[z*1088 + h*17 + f] : 0.0f;
  }
  for (int i = tid; i < 4096; i += 128) {
    int g = i >> 6, h = i & 63;
    sm[OW1 + g*65 + h] = W1[z*4096 + i];
    sm[OW2 + g*65 + h] = W2[z*4096 + i];
  }
  if (tid < 64) {
    sm[OW3 + tid] = W3[z*64 + tid];
    sm[OB0 + tid] = b0[z*64 + tid];
    sm[OB1 + tid] = b1[z*64 + tid];
    sm[OB2 + tid] = b2[z*64 + tid];
  }
  if (tid == 0) sm[OB3] = b3[z];
  __syncthreads();

  float* scr = &sm[OSCR + wave * SCR_PER_WAVE];

  /* input window tile -> scratch rows [n_local][f0..19] */
  for (int e = lane; e < 16*20; e += 32) {
    int nl = e / 20, f = e % 20;
    int n = n0 + nl;
    int b = n / TW;
    int t = n - b * TW;
    float v = 0.0f;
    if (f < 16)       v = x[((size_t)(b*TDIM + t))*16 + f];
    else if (f == 16) v = x[((size_t)(b*TDIM + t + 1))*16 + z];
    scr[nl*SCR_STRIDE + f] = v;
  }

  v8f acc[4];
  unsigned d0m = 0, d1m = 0, d2m = 0;

  /* ---- layer 0: p0 = inp(16x20) * W0pad^T, B[k=f][col=h] = sW0[col*20+f] ---- */
  #pragma unroll
  for (int t4 = 0; t4 < 4; ++t4) acc[t4] = splat8(sm[OB0 + t4*16 + ln]);
  for (int kk = 0; kk < 5; ++kk) {
    int k0 = kk*4 + kh2;
    const float* ap = scr + ln*SCR_STRIDE + k0;
    v2f a; a.x = ap[0]; a.y = ap[1];
    #pragma unroll
    for (int t4 = 0; t4 < 4; ++t4) {
      const float* bp = &sm[OW0 + (t4*16 + ln)*20 + k0];
      v2f b; b.x = bp[0]; b.y = bp[1];
      acc[t4] = wmma4(a, b, acc[t4]);
    }
  }
  #pragma unroll
  for (int t4 = 0; t4 < 4; ++t4)
    #pragma unroll
    for (int r = 0; r < 8; ++r) {
      float p = acc[t4][r]; bool pos = p >= 0.0f;
      d0m |= (pos ? 1u : 0u) << (t4*8 + r);
      scr[(r + 8*half)*SCR_STRIDE + t4*16 + ln] = pos ? p : SLOPEV*p;
    }

  /* ---- layer 1 ---- */
  gemm_fwd64(scr, &sm[OW1], &sm[OB1], ln, kh2, acc);
  #pragma unroll
  for (int t4 = 0; t4 < 4; ++t4)
    #pragma unroll
    for (int r = 0; r < 8; ++r) {
      float p = acc[t4][r]; bool pos = p >= 0.0f;
      d1m |= (pos ? 1u : 0u) << (t4*8 + r);
      scr[(r + 8*half)*SCR_STRIDE + t4*16 + ln] = pos ? p : SLOPEV*p;
    }

  /* ---- layer 2 + residual head ---- */
  gemm_fwd64(scr, &sm[OW2], &sm[OB2], ln, kh2, acc);
  v8f partial = splat8(0.0f);
  #pragma unroll
  for (int t4 = 0; t4 < 4; ++t4) {
    float w3t = sm[OW3 + t4*16 + ln];
    #pragma unroll
    for (int r = 0; r < 8; ++r) {
      float p = acc[t4][r]; bool pos = p >= 0.0f;
      d2m |= (pos ? 1u : 0u) << (t4*8 + r);
      partial[r] += (pos ? p : SLOPEV*p) * w3t;
    }
  }
  #pragma unroll
  for (int m = 8; m >= 1; m >>= 1)
    #pragma unroll
    for (int r = 0; r < 8; ++r) partial[r] += __shfl_xor(partial[r], m, 32);
  if (ln == 0) {
    float bb3 = sm[OB3];
    #pragma unroll
    for (int r = 0; r < 8; ++r)
      out[RES_OFF + (size_t)z*NTOT + n0 + 8*half + r] = partial[r] + bb3;
  }

  /* ---- backward chain ---- */
  #pragma unroll
  for (int t4 = 0; t4 < 4; ++t4) {
    float w3t = sm[OW3 + t4*16 + ln];
    #pragma unroll
    for (int r = 0; r < 8; ++r) {
      float d = ((d2m >> (t4*8 + r)) & 1u) ? 1.0f : SLOPEV;
      scr[(r + 8*half)*SCR_STRIDE + t4*16 + ln] = w3t * d;
    }
  }
  gemm_bwd64(scr, &sm[OW2], ln, kh2, acc);
  #pragma unroll
  for (int t4 = 0; t4 < 4; ++t4)
    #pragma unroll
    for (int r = 0; r < 8; ++r) {
      float d = ((d1m >> (t4*8 + r)) & 1u) ? 1.0f : SLOPEV;
      scr[(r + 8*half)*SCR_STRIDE + t4*16 + ln] = acc[t4][r] * d;
    }
  gemm_bwd64(scr, &sm[OW1], ln, kh2, acc);
  #pragma unroll
  for (int t4 = 0; t4 < 4; ++t4)
    #pragma unroll
    for (int r = 0; r < 8; ++r) {
      float d = ((d0m >> (t4*8 + r)) & 1u) ? 1.0f : SLOPEV;
      scr[(r + 8*half)*SCR_STRIDE + t4*16 + ln] = acc[t4][r] * d;
    }

  /* jac = v0 * W0 : B[k=h][col=f] = sW0[k*20+f]; tile1 col0 = f=16 */
  v8f jacL = splat8(0.0f), jacR = splat8(0.0f);
  for (int kk = 0; kk < 16; ++kk) {
    int k0 = kk*4 + kh2;
    const float* ap = scr + ln*SCR_STRIDE + k0;
    v2f a; a.x = ap[0]; a.y = ap[1];
    v2f bL; bL.x = sm[OW0 + k0*20 + ln]; bL.y = sm[OW0 + (k0+1)*20 + ln];
    jacL = wmma4(a, bL, jacL);
    int f = 16 + ln;
    v2f bR;
    bR.x = (f < 20) ? sm[OW0 + k0*20 + f] : 0.0f;
    bR.y = (f < 20) ? sm[OW0 + (k0+1)*20 + f] : 0.0f;
    jacR = wmma4(a, bR, jacR);
  }

  /* sld: log|jac16+eps| summed into out[SLD_OFF + b] */
  if (ln == 0) {
    float accum = 0.0f;
    int bprev = (n0 + 8*half) / TW;
    #pragma unroll
    for (int r = 0; r < 8; ++r) {
      int n = n0 + 8*half + r;
      int b = n / TW;
      float l = logf(fabsf(jacR[r] + EPSV));
      if (b != bprev) { atomicAdd(&out[SLD_OFF + bprev], accum); accum = 0.0f; bprev = b; }
      accum += l;
    }
    atomicAdd(&out[SLD_OFF + bprev], accum);
  }

  /* tj = -jacL / (jac16 + eps), flat index n*256 + z*16 + f */
  #pragma unroll
  for (int r = 0; r < 8; ++r) {
    float j16 = __shfl(jacR[r], lane & 16, 32);
    float tjv = -jacL[r] / (j16 + EPSV);
    int n = n0 + 8*half + r;
    out[TJ_OFF + (size_t)n*256 + z*16 + ln] = tjv;
  }
}

__global__ void zero_sld_kernel(float* __restrict__ out) {
  if (threadIdx.x < BDIM) out[SLD_OFF + threadIdx.x] = 0.0f;
}

/* per-t nonzero-count means over tj: xx_spa[t] and yy_spa[t-1] are identical values */
__global__ __launch_bounds__(256)
void spa_kernel(float* __restrict__ out) {
  const int t = blockIdx.x;      /* 0..TW-1 */
  const int tid = threadIdx.x;   /* 0..255 covers z*16+f */
  float cnt = 0.0f;
  for (int b = 0; b < BDIM; ++b) {
    float v = out[TJ_OFF + (size_t)(b*TW + t)*256 + tid];
    cnt += (v != 0.0f) ? 1.0f : 0.0f;
  }
  __shared__ float red[256];
  red[tid] = cnt;
  __syncthreads();
  for (int s = 128; s > 0; s >>= 1) {
    if (tid < s) red[tid] += red[tid + s];
    __syncthreads();
  }
  if (tid == 0) {
    float m = red[0] / 2048.0f;
    out[XX_OFF + t] = m;
    if (t > 0) out[YY_OFF + (t - 1)] = m;
  }
}

extern "C" void kernel_launch(void* const* d_in, const int* in_sizes, int n_in,
                              void* d_out, int out_size, void* d_ws, size_t ws_size,
                              hipStream_t stream) {
  (void)in_sizes; (void)n_in; (void)out_size; (void)d_ws; (void)ws_size;
  const float* x  = (const float*)d_in[0];
  const float* W0 = (const float*)d_in[1];
  const float* b0 = (const float*)d_in[2];
  const float* W1 = (const float*)d_in[3];
  const float* b1 = (const float*)d_in[4];
  const float* W2 = (const float*)d_in[5];
  const float* b2 = (const float*)d_in[6];
  const float* W3 = (const float*)d_in[7];
  const float* b3 = (const float*)d_in[8];
  float* out = (float*)d_out;

  zero_sld_kernel<<<dim3(1), dim3(128), 0, stream>>>(out);
  mlp_jac_kernel<<<dim3(16*2046), dim3(128), 0, stream>>>(x, W0, b0, W1, b1, W2, b2, W3, b3, out);
  spa_kernel<<<dim3(TW), dim3(256), 0, stream>>>(out);
}